// SURFMNetLoss_49873160241229
// MI455X (gfx1250) — compile-verified
//
#include <hip/hip_runtime.h>
#include <stdint.h>

typedef float v2f __attribute__((ext_vector_type(2)));
typedef float v8f __attribute__((ext_vector_type(8)));
typedef unsigned int u32x4 __attribute__((ext_vector_type(4)));
typedef int i32x8 __attribute__((ext_vector_type(8)));
typedef int i32x4 __attribute__((ext_vector_type(4)));

#define B_      8
#define N_      6000
#define K_      120
#define KP      128
#define F_      352
#define ND_     70
#define KC      48
#define NCHUNK  125      // 6000 / 48

// LDS layout strides (floats)
#define ETS     51       // 48 data + 1 pad per 16 DWORDs (TDM pad encoding), gcd(51,64)=1
#define GTS     52       // transposed G stride: even (b64-aligned), conflict-free frags

// phase-1 LDS float offsets (within 32768-float dynamic LDS)
#define ET1_OF  0
#define ET2_OF  6528     // 128*51
#define GT1_OF  13056
#define GT2_OF  19712    // + 128*52 = 26368 total

// workspace layout (float offsets)
#define PRE_OFS  0        // 560 per-(a,b) partials
#define BIJ_OFS  560      // 8
#define ORTH_OFS 568      // 8
#define LAP_OFS  576      // 8
#define F1_OFS   1024                       // 8*70*6000
#define F2_OFS   (1024 + B_ * ND_ * N_)

static __device__ __forceinline__ v8f wmma_f32(v2f a, v2f b, v8f c) {
  // D = A(16x4,f32) * B(4x16,f32) + C(16x16,f32)
  return __builtin_amdgcn_wmma_f32_16x16x4_f32(
      false, a, false, b, (short)0, c, false, false);
}

#if defined(__has_builtin)
#if __has_builtin(__builtin_amdgcn_tensor_load_to_lds) && \
    __has_builtin(__builtin_amdgcn_s_wait_tensorcnt)
#define USE_TDM 1
#endif
#endif

#ifdef USE_TDM
// 2D TDM load: tile 48 floats x 120 rows from a row-major [120][6000] tensor,
// LDS padding: +1 DWORD after every 16 DWORDs -> LDS row stride 51 floats.
static __device__ __forceinline__ void tdm_load_et(uint32_t lds_byte_off,
                                                   const float* tile_start) {
  uint64_t ga = (uint64_t)(uintptr_t)tile_start;
  u32x4 g0;
  g0[0] = 1u;                                  // count=1, user descriptor
  g0[1] = lds_byte_off;                        // lds_addr
  g0[2] = (uint32_t)ga;                        // global_addr[31:0]
  g0[3] = (uint32_t)(ga >> 32) | 0x80000000u;  // global_addr[56:32] | type=2<<30
  i32x8 g1;
  g1[0] = (2 << 16) | (1 << 20) | (3 << 22);   // data_size=4B, pad_en, interval=16DW, amount=1DW
  g1[1] = KC << 16;                            // tensor_dim0[15:0] = 48
  g1[2] = K_ << 16;                            // tensor_dim0 hi=0 | tensor_dim1 lo = 120
  g1[3] = KC << 16;                            // tensor_dim1 hi=0 | tile_dim0 = 48
  g1[4] = K_;                                  // tile_dim1 = 120, tile_dim2 = 0
  g1[5] = N_;                                  // tensor_dim0_stride = 6000
  g1[6] = 0;
  g1[7] = 0;
  i32x4 z4 = {};
  i32x8 z8 = {};
  // amdgpu-toolchain (clang-23) 6-arg form: (g0, g1, g2, g3, g4, cpol)
  __builtin_amdgcn_tensor_load_to_lds(g0, g1, z4, z4, z8, 0);
}
#endif

// ---------------------------------------------------------------------------
// K0: gather feat[:, :, descs] transposed -> f1t[a][b][n] (contiguous in n)
// ---------------------------------------------------------------------------
__global__ void gather_kernel(const float* __restrict__ feat1,
                              const float* __restrict__ feat2,
                              const long long* __restrict__ descs,
                              float* __restrict__ ws) {
  int n = blockIdx.x * blockDim.x + threadIdx.x;
  if (n >= N_) return;
  int b = blockIdx.y;
  int z = blockIdx.z;
  int a = z & 7, s = z >> 3;
  int d = (int)descs[b];
  const float* feat = s ? feat2 : feat1;
  float v = feat[((size_t)a * N_ + n) * F_ + d];
  ws[(s ? F2_OFS : F1_OFS) + ((size_t)a * ND_ + b) * N_ + n] = v;
}

// ---------------------------------------------------------------------------
// K1: per (a,b): M1 = ET1 diag(f1) E1, M2 = ET2 diag(f2) E2   (120x6000x120)
//     then sum((C1 M1 - M2 C1)^2) + sum((C2 M2 - M1 C2)^2) -> partial
// 256 threads = 8 wave32; wave w owns output rows [16w,16w+16)
// ---------------------------------------------------------------------------
__global__ void __launch_bounds__(256) einsum_pre_kernel(
    const float* __restrict__ C1g, const float* __restrict__ C2g,
    const float* __restrict__ E1g, const float* __restrict__ E2g,
    const float* __restrict__ ET1g, const float* __restrict__ ET2g,
    float* __restrict__ ws) {
  extern __shared__ float smem[];  // 32768 floats = 128 KB dynamic LDS
  const int wg = blockIdx.x;
  const int a = wg / ND_, b = wg % ND_;
  const int tid = threadIdx.x;
  const int w = tid >> 5;          // wave id 0..7
  const int l = tid & 31;
  const int h = l >> 4;            // lane half (K pairing for 16x16x4)
  const int r = l & 15;

  const float* c1  = C1g + a * K_ * K_;
  const float* c2  = C2g + a * K_ * K_;
  const float* e1  = E1g + (size_t)a * N_ * K_;
  const float* e2  = E2g + (size_t)a * N_ * K_;
  const float* et1 = ET1g + (size_t)a * K_ * N_;
  const float* et2 = ET2g + (size_t)a * K_ * N_;
  const float* f1  = ws + F1_OFS + ((size_t)a * ND_ + b) * N_;
  const float* f2  = ws + F2_OFS + ((size_t)a * ND_ + b) * N_;

  float* ET1c = smem + ET1_OF;   // [row f][51], rows 120..127 zero
  float* ET2c = smem + ET2_OF;
  float* Gt1  = smem + GT1_OF;   // transposed: [col e][52]
  float* Gt2  = smem + GT2_OF;

  // zero the pad rows of ET (f = 120..127) once; never rewritten in the loop
  for (int i = tid; i < 8 * ETS; i += 256) {
    ET1c[K_ * ETS + i] = 0.f;
    ET2c[K_ * ETS + i] = 0.f;
  }

  v8f acc1[8] = {}, acc2[8] = {};

  for (int it = 0; it < NCHUNK; ++it) {
    const int n0 = it * KC;
#ifdef USE_TDM
    if (w == 0) {  // one wave drives the Tensor Data Mover for both ET tiles
      tdm_load_et((uint32_t)(uintptr_t)ET1c, et1 + n0);
      tdm_load_et((uint32_t)(uintptr_t)ET2c, et2 + n0);
    }
#else
    for (int i = tid; i < K_ * KC; i += 256) {
      int f = i / KC, k = i - f * KC;
      ET1c[f * ETS + k + (k >> 4)] = et1[(size_t)f * N_ + n0 + k];
      ET2c[f * ETS + k + (k >> 4)] = et2[(size_t)f * N_ + n0 + k];
    }
#endif
    // stage Gt[e][k] = f[n0+k] * E[n0+k][e]  (cols 120..127 zero)
    for (int i = tid; i < KC * KP; i += 256) {
      int k = i >> 7, e = i & 127;                 // e fastest: coalesced E reads
      float fa = f1[n0 + k], fb = f2[n0 + k];
      float va = 0.f, vb = 0.f;
      if (e < K_) {
        va = fa * e1[(size_t)(n0 + k) * K_ + e];
        vb = fb * e2[(size_t)(n0 + k) * K_ + e];
      }
      Gt1[e * GTS + k] = va;
      Gt2[e * GTS + k] = vb;
    }
    if (it + 1 < NCHUNK) {
      __builtin_prefetch(f1 + n0 + KC, 0, 1);   // global_prefetch_b8
      __builtin_prefetch(f2 + n0 + KC, 0, 1);
    }
#ifdef USE_TDM
    if (w == 0) __builtin_amdgcn_s_wait_tensorcnt(0);
#endif
    __syncthreads();

    #pragma unroll
    for (int t = 0; t < KC / 4; ++t) {
      const int krow = 4 * t + 2 * h;
      const int pidx = (16 * w + r) * ETS + krow + (krow >> 4);  // skip pads
      v2f a1, a2;
      a1.x = ET1c[pidx]; a1.y = ET1c[pidx + 1];
      a2.x = ET2c[pidx]; a2.y = ET2c[pidx + 1];
      #pragma unroll
      for (int c = 0; c < 8; ++c) {
        v2f b1 = *(const v2f*)(Gt1 + (16 * c + r) * GTS + krow);  // 8B aligned
        v2f b2 = *(const v2f*)(Gt2 + (16 * c + r) * GTS + krow);
        acc1[c] = wmma_f32(a1, b1, acc1[c]);
        acc2[c] = wmma_f32(a2, b2, acc2[c]);
      }
    }
    __syncthreads();
  }

  // spill M1/M2 (zero-padded 128x128) into LDS
  float* M1s = smem;           // 16384
  float* M2s = smem + 16384;   // 16384
  #pragma unroll
  for (int c = 0; c < 8; ++c) {
    #pragma unroll
    for (int v = 0; v < 8; ++v) {
      int row = 16 * w + 8 * h + v;
      int col = 16 * c + r;
      M1s[row * KP + col] = acc1[c][v];
      M2s[row * KP + col] = acc2[c][v];
    }
  }
  __syncthreads();

  // phase 2: src = Cg @ Msrc (LDS), tgt = Mtgt (LDS) @ Cg ; sum (src-tgt)^2
  float ssum = 0.f;
  const int row = 16 * w + r;
  const bool rok = row < K_;
  #pragma unroll
  for (int p = 0; p < 2; ++p) {
    const float* cg = p ? c2 : c1;
    const float* Msrc = p ? M2s : M1s;
    const float* Mtgt = p ? M1s : M2s;
    v8f S[8] = {}, T[8] = {};
    for (int t = 0; t < 32; ++t) {
      const int kA = 4 * t + 2 * h;
      const bool kok = kA < K_;          // kA even => kA+1<K_ iff kA<K_
      v2f aS, aT;
      aS.x = (rok && kok) ? cg[row * K_ + kA] : 0.f;
      aS.y = (rok && kok) ? cg[row * K_ + kA + 1] : 0.f;
      aT = *(const v2f*)(Mtgt + row * KP + kA);
      #pragma unroll
      for (int c = 0; c < 8; ++c) {
        const int col = 16 * c + r;
        const bool cok = col < K_;
        v2f bS, bT;
        bS.x = Msrc[kA * KP + col];
        bS.y = Msrc[(kA + 1) * KP + col];
        bT.x = (kok && cok) ? cg[kA * K_ + col] : 0.f;
        bT.y = (kok && cok) ? cg[(kA + 1) * K_ + col] : 0.f;
        S[c] = wmma_f32(aS, bS, S[c]);
        T[c] = wmma_f32(aT, bT, T[c]);
      }
    }
    #pragma unroll
    for (int c = 0; c < 8; ++c)
      #pragma unroll
      for (int v = 0; v < 8; ++v) {
        float d = S[c][v] - T[c][v];
        ssum += d * d;   // padded region is exactly 0-0
      }
  }

  // block reduce -> one partial per (a,b)
  __syncthreads();
  smem[tid] = ssum;
  __syncthreads();
  for (int s2 = 128; s2 > 0; s2 >>= 1) {
    if (tid < s2) smem[tid] += smem[tid + s2];
    __syncthreads();
  }
  if (tid == 0) ws[PRE_OFS + wg] = smem[0];
}

// ---------------------------------------------------------------------------
// K2: per a: bij = |C1C2-I|^2+|C2C1-I|^2, orth = |C1^T C1-I|^2+|C2^T C2-I|^2,
//            lap elementwise term
// ---------------------------------------------------------------------------
__global__ void __launch_bounds__(256) small_terms_kernel(
    const float* __restrict__ C1g, const float* __restrict__ C2g,
    const float* __restrict__ ev1g, const float* __restrict__ ev2g,
    float* __restrict__ ws) {
  extern __shared__ float smem[];  // 32768 floats
  const int a = blockIdx.x;
  const int tid = threadIdx.x;
  const int w = tid >> 5, l = tid & 31, h = l >> 4, r = l & 15;
  const float* c1 = C1g + a * K_ * K_;
  const float* c2 = C2g + a * K_ * K_;
  const float* ev1 = ev1g + a * K_;
  const float* ev2 = ev2g + a * K_;
  float* C1s = smem;            // 16384 (padded 128x128)
  float* C2s = smem + 16384;

  for (int i = tid; i < KP * KP; i += 256) {
    int row = i >> 7, col = i & 127;
    bool ok = (row < K_) && (col < K_);
    C1s[i] = ok ? c1[row * K_ + col] : 0.f;
    C2s[i] = ok ? c2[row * K_ + col] : 0.f;
  }
  __syncthreads();

  float bs = 0.f, os = 0.f, ls = 0.f;

  // q=0: C1@C2   q=1: C2@C1   q=2: C1^T@C1   q=3: C2^T@C2
  #pragma unroll
  for (int q = 0; q < 4; ++q) {
    const float* As = (q == 0 || q == 2) ? C1s : C2s;
    const float* Bs = (q == 0) ? C2s : (q == 1) ? C1s : As;
    const bool transA = (q >= 2);
    v8f Acc[8] = {};
    for (int t = 0; t < 32; ++t) {
      const int kA = 4 * t + 2 * h;
      v2f av;
      if (transA) {
        av.x = As[kA * KP + 16 * w + r];
        av.y = As[(kA + 1) * KP + 16 * w + r];
      } else {
        av = *(const v2f*)(As + (16 * w + r) * KP + kA);
      }
      #pragma unroll
      for (int c = 0; c < 8; ++c) {
        v2f bv;
        bv.x = Bs[kA * KP + 16 * c + r];
        bv.y = Bs[(kA + 1) * KP + 16 * c + r];
        Acc[c] = wmma_f32(av, bv, Acc[c]);
      }
    }
    float sq = 0.f;
    #pragma unroll
    for (int c = 0; c < 8; ++c)
      #pragma unroll
      for (int v = 0; v < 8; ++v) {
        int row = 16 * w + 8 * h + v, col = 16 * c + r;
        float x = Acc[c][v] - ((row == col && row < K_) ? 1.f : 0.f);
        sq += x * x;
      }
    if (q < 2) bs += sq; else os += sq;
  }

  // laplacian term
  for (int i = tid; i < K_ * K_; i += 256) {
    int row = i / K_, col = i - row * K_;
    float d1 = c1[i] * (ev1[col] - ev2[row]);
    float d2 = c2[i] * (ev2[col] - ev1[row]);
    ls += d1 * d1 + d2 * d2;
  }

  float vals[3] = {bs, os, ls};
  const int ofs[3] = {BIJ_OFS, ORTH_OFS, LAP_OFS};
  for (int q = 0; q < 3; ++q) {
    __syncthreads();
    smem[tid] = vals[q];
    __syncthreads();
    for (int s2 = 128; s2 > 0; s2 >>= 1) {
      if (tid < s2) smem[tid] += smem[tid + s2];
      __syncthreads();
    }
    if (tid == 0) ws[ofs[q] + a] = smem[0];
  }
}

// ---------------------------------------------------------------------------
// K3: deterministic final reduction + weighting
// ---------------------------------------------------------------------------
__global__ void __launch_bounds__(256) finalize_kernel(
    const float* __restrict__ ws, float* __restrict__ out) {
  __shared__ float red[256];
  const int tid = threadIdx.x;
  float s = 0.f;
  for (int i = tid; i < B_ * ND_; i += 256) s += ws[PRE_OFS + i];
  red[tid] = s;
  __syncthreads();
  for (int st = 128; st > 0; st >>= 1) {
    if (tid < st) red[tid] += red[tid + st];
    __syncthreads();
  }
  if (tid == 0) {
    float pre = red[0];
    float bij = 0.f, orth = 0.f, lap = 0.f;
    for (int a = 0; a < B_; ++a) {
      bij += ws[BIJ_OFS + a];
      orth += ws[ORTH_OFS + a];
      lap += ws[LAP_OFS + a];
    }
    out[0] = 1000.f * (bij / 8.f) + 1000.f * (orth / 8.f) + (lap / 8.f) +
             100000.f * (pre / (8.f * 120.f * 120.f));
  }
}

// ---------------------------------------------------------------------------
extern "C" void kernel_launch(void* const* d_in, const int* in_sizes, int n_in,
                              void* d_out, int out_size, void* d_ws, size_t ws_size,
                              hipStream_t stream) {
  (void)in_sizes; (void)n_in; (void)out_size; (void)ws_size;
  const float* C1    = (const float*)d_in[0];
  const float* C2    = (const float*)d_in[1];
  const float* feat1 = (const float*)d_in[2];
  const float* feat2 = (const float*)d_in[3];
  const float* E1    = (const float*)d_in[4];
  const float* E2    = (const float*)d_in[5];
  const float* ET1   = (const float*)d_in[6];
  const float* ET2   = (const float*)d_in[7];
  const float* ev1   = (const float*)d_in[8];
  const float* ev2   = (const float*)d_in[9];
  const long long* descs = (const long long*)d_in[10];
  float* ws  = (float*)d_ws;
  float* out = (float*)d_out;

  dim3 g0((N_ + 255) / 256, ND_, 16);
  gather_kernel<<<g0, dim3(256), 0, stream>>>(feat1, feat2, descs, ws);
  einsum_pre_kernel<<<B_ * ND_, 256, 131072, stream>>>(C1, C2, E1, E2, ET1, ET2, ws);
  small_terms_kernel<<<B_, 256, 131072, stream>>>(C1, C2, ev1, ev2, ws);
  finalize_kernel<<<1, 256, 0, stream>>>(ws, out);
}